// LSTM_15736760172713
// MI455X (gfx1250) — compile-verified
//
#include <hip/hip_runtime.h>
#include <math.h>

// LSTM: T=512, B=64, I=128, H=512, L=2, O=256. fp32 throughout, WMMA f32 16x16x4.

typedef float v2f __attribute__((ext_vector_type(2)));
typedef float v8f __attribute__((ext_vector_type(8)));

#define TT 512
#define BB 64
#define II 128
#define HH 512
#define G4 2048   // 4*H
#define OO 256
#define NWG_REC 32

__device__ __forceinline__ float sigmoidf_(float x) {
    return 1.0f / (1.0f + __expf(-x));
}

// Grid-wide barrier across NWG_REC workgroups using a monotonic counter.
__device__ __forceinline__ void grid_sync(unsigned* ctr, unsigned target) {
    __threadfence();
    __syncthreads();
    if (threadIdx.x == 0) {
        __hip_atomic_fetch_add(ctr, 1u, __ATOMIC_ACQ_REL, __HIP_MEMORY_SCOPE_AGENT);
        while (__hip_atomic_load(ctr, __ATOMIC_ACQUIRE, __HIP_MEMORY_SCOPE_AGENT) < target) {
            __builtin_amdgcn_s_sleep(2);
        }
    }
    __syncthreads();
    __threadfence();
}

// Batched transpose: in[batch][R][C] -> out[batch][C][R]
__global__ void k_transpose(const float* __restrict__ in, float* __restrict__ out,
                            int R, int C) {
    size_t base = (size_t)blockIdx.y * R * C;
    int n = R * C;
    for (int idx = blockIdx.x * blockDim.x + threadIdx.x; idx < n;
         idx += gridDim.x * blockDim.x) {
        int r = idx / C, c = idx - r * C;
        out[base + (size_t)c * R + r] = in[base + idx];
    }
}

// C[M][N] = A @ W^T + bias.
// A stored blocked K-major: [M/64][K][64];  WT stored [K][N];  out row-major [M][N].
// 256 threads = 8 wave32; block computes a 64x64 C tile; each wave: 2 16x16 tiles.
__global__ __launch_bounds__(256) void k_gemm_wmma(
    const float* __restrict__ A, const float* __restrict__ WT,
    const float* __restrict__ bias, float* __restrict__ out, int K, int N) {
    int nblk = N >> 6;
    int mb = blockIdx.x / nblk;
    int nb = blockIdx.x - mb * nblk;
    int w = threadIdx.x >> 5, lane = threadIdx.x & 31;
    int l16 = lane & 15, hi = lane >> 4;
    int mt = w & 3, np = w >> 2;
    int arow = mt * 16 + l16;
    int gc0 = nb * 64 + np * 32 + l16;
    int gc1 = gc0 + 16;
    const float* Ab = A + (size_t)mb * K * 64;
    int kb = hi << 1;  // lanes 0-15: K+0/K+1, lanes 16-31: K+2/K+3

    v8f acc0 = {0.f,0.f,0.f,0.f,0.f,0.f,0.f,0.f};
    v8f acc1 = {0.f,0.f,0.f,0.f,0.f,0.f,0.f,0.f};
    for (int k = 0; k < K; k += 4) {
        v2f a, bv0, bv1;
        const float* Ak = Ab + (size_t)(k + kb) * 64;
        a.x = Ak[arow];
        a.y = Ak[64 + arow];
        const float* Wk = WT + (size_t)(k + kb) * N;
        bv0.x = Wk[gc0]; bv0.y = Wk[N + gc0];
        bv1.x = Wk[gc1]; bv1.y = Wk[N + gc1];
        acc0 = __builtin_amdgcn_wmma_f32_16x16x4_f32(false, a, false, bv0,
                                                     (short)0, acc0, false, false);
        acc1 = __builtin_amdgcn_wmma_f32_16x16x4_f32(false, a, false, bv1,
                                                     (short)0, acc1, false, false);
    }
    float bb0 = bias[gc0], bb1 = bias[gc1];
    size_t mbase = (size_t)mb * 64 + mt * 16 + hi * 8;
#pragma unroll
    for (int v = 0; v < 8; ++v) {
        size_t r = (mbase + v) * (size_t)N;
        out[r + gc0] = acc0[v] + bb0;
        out[r + gc1] = acc1[v] + bb1;
    }
}

// Persistent recurrence kernel: 32 WGs x 256 threads. WG j owns h-cols [16j,16j+16).
// Per step: gates[64 b x 64 cols] = xg_t + h_{t-1} @ Whh^T (WMMA), then cell.
// hbuf: double-buffered h, K-major [2][512][64]. yT out: [T][512][64].
__global__ __launch_bounds__(256) void k_lstm_rec(
    const float* __restrict__ xg,   // [T*64][2048] gates input contribution (+bias)
    const float* __restrict__ WhT,  // [512][2048]  (K-major transposed Whh)
    float* __restrict__ hbuf,       // [2][512*64]
    float* __restrict__ yT,         // [T][512][64]
    unsigned* __restrict__ ctr,
    float* __restrict__ hN,         // [64][512]
    float* __restrict__ cN) {       // [64][512]
    __shared__ float g_lds[64 * 65];  // padded: stride 65 kills bank conflicts
    const int j = blockIdx.x;
    const int w = threadIdx.x >> 5, lane = threadIdx.x & 31;
    const int l16 = lane & 15, hi = lane >> 4;
    const int mt = w & 3, np = w >> 2;
    const int g0 = np * 2, g1 = g0 + 1;                 // gate tiles (i,f | g,o)
    const int arow = mt * 16 + l16;
    const int gc0 = g0 * 512 + j * 16 + l16;            // global gate columns
    const int gc1 = g1 * 512 + j * 16 + l16;
    const int kb = hi << 1;
    float cst[4] = {0.f, 0.f, 0.f, 0.f};                // c-state, 4 elems/thread

    for (int t = 0; t < TT; ++t) {
        const float* hr = hbuf + (size_t)(t & 1) * (HH * BB);
        float* hw = hbuf + (size_t)((t + 1) & 1) * (HH * BB);

        // acc = xg tile (bias already folded in by k_gemm_wmma)
        v8f acc0, acc1;
        const float* xgt = xg + (size_t)t * BB * G4;
        {
            size_t rbase = (size_t)(mt * 16 + hi * 8);
#pragma unroll
            for (int v = 0; v < 8; ++v) {
                size_t r = (rbase + v) * G4;
                acc0[v] = xgt[r + gc0];
                acc1[v] = xgt[r + gc1];
            }
        }
        // += h_{t-1} @ Whh^T  (h0 == 0 -> skip at t==0)
        if (t > 0) {
            for (int k = 0; k < HH; k += 4) {
                v2f a, bv0, bv1;
                const float* hk = hr + (k + kb) * 64;
                a.x = hk[arow];
                a.y = hk[64 + arow];
                const float* Wk = WhT + (size_t)(k + kb) * G4;
                bv0.x = Wk[gc0]; bv0.y = Wk[G4 + gc0];
                bv1.x = Wk[gc1]; bv1.y = Wk[G4 + gc1];
                acc0 = __builtin_amdgcn_wmma_f32_16x16x4_f32(false, a, false, bv0,
                                                             (short)0, acc0, false, false);
                acc1 = __builtin_amdgcn_wmma_f32_16x16x4_f32(false, a, false, bv1,
                                                             (short)0, acc1, false, false);
            }
        }
        // scatter gate tiles to LDS: [b][gate*16+m]
        {
            int c0 = g0 * 16 + l16, c1 = g1 * 16 + l16;
            int rb = mt * 16 + hi * 8;
#pragma unroll
            for (int v = 0; v < 8; ++v) {
                g_lds[(rb + v) * 65 + c0] = acc0[v];
                g_lds[(rb + v) * 65 + c1] = acc1[v];
            }
        }
        __syncthreads();
        // LSTM cell: 1024 (b,m) elems / 256 threads
#pragma unroll
        for (int i = 0; i < 4; ++i) {
            int e = (int)threadIdx.x + (i << 8);
            int b = e & 63, m = e >> 6;
            float ig = sigmoidf_(g_lds[b * 65 + m]);
            float fg = sigmoidf_(g_lds[b * 65 + 16 + m]);
            float gg = tanhf(g_lds[b * 65 + 32 + m]);
            float og = sigmoidf_(g_lds[b * 65 + 48 + m]);
            float c = fg * cst[i] + ig * gg;
            cst[i] = c;
            float h = og * tanhf(c);
            int krow = j * 16 + m;
            hw[krow * 64 + b] = h;
            yT[(size_t)t * (HH * BB) + krow * 64 + b] = h;
            if (t == TT - 1) {
                hN[b * HH + krow] = h;
                cN[b * HH + krow] = c;
            }
        }
        grid_sync(ctr, (unsigned)(t + 1) * NWG_REC);
    }
}

extern "C" void kernel_launch(void* const* d_in, const int* in_sizes, int n_in,
                              void* d_out, int out_size, void* d_ws, size_t ws_size,
                              hipStream_t stream) {
    (void)in_sizes; (void)n_in; (void)out_size; (void)ws_size;
    const float* x    = (const float*)d_in[0];
    const float* Wih0 = (const float*)d_in[1];
    const float* Whh0 = (const float*)d_in[2];
    const float* b0   = (const float*)d_in[3];
    const float* Wih1 = (const float*)d_in[4];
    const float* Whh1 = (const float*)d_in[5];
    const float* b1   = (const float*)d_in[6];
    const float* fcw  = (const float*)d_in[7];
    const float* fcb  = (const float*)d_in[8];

    float* out = (float*)d_out;
    float* fc  = out;                                  // [T*B][O]
    float* hN  = out + (size_t)TT * BB * OO;           // [2][B][H]
    float* cN  = hN + (size_t)2 * BB * HH;

    float* ws = (float*)d_ws;
    unsigned* ctr = (unsigned*)ws;                     // 2 counters, 256B reserved
    size_t off = 64;
    float* xg   = ws + off; off += (size_t)TT * BB * G4;  // 256 MB
    float* xT   = ws + off; off += (size_t)TT * II * BB;  // 16 MB  [T][I][B]
    float* WTa  = ws + off; off += (size_t)HH * G4;       // 4 MB   input-weight^T
    float* WTh  = ws + off; off += (size_t)HH * G4;       // 4 MB   Whh^T
    float* WTfc = ws + off; off += (size_t)HH * OO;       // 0.5 MB fc_w^T
    float* hbuf = ws + off; off += (size_t)2 * HH * BB;   // double-buffered h
    float* y0t  = ws + off; off += (size_t)TT * HH * BB;  // 64 MB  [T][H][B]
    float* y1t  = ws + off; off += (size_t)TT * HH * BB;  // 64 MB

    hipMemsetAsync(ctr, 0, 256, stream);

    dim3 blk(256);
    const int gemm_blocks_g4 = (TT * BB / 64) * (G4 / 64);  // 16384
    const int gemm_blocks_fc = (TT * BB / 64) * (OO / 64);  // 2048

    // ---- Layer 0 ----
    k_transpose<<<dim3(32, TT), blk, 0, stream>>>(x, xT, BB, II);        // x -> [T][I][B]
    k_transpose<<<dim3(1024, 1), blk, 0, stream>>>(Wih0, WTa, G4, II);   // -> [I][4H]
    k_gemm_wmma<<<dim3(gemm_blocks_g4), blk, 0, stream>>>(xT, WTa, b0, xg, II, G4);
    k_transpose<<<dim3(4096, 1), blk, 0, stream>>>(Whh0, WTh, G4, HH);   // -> [H][4H]
    k_lstm_rec<<<dim3(NWG_REC), blk, 0, stream>>>(xg, WTh, hbuf, y0t, ctr + 0,
                                                  hN, cN);
    // ---- Layer 1 ----
    k_transpose<<<dim3(4096, 1), blk, 0, stream>>>(Wih1, WTa, G4, HH);
    k_gemm_wmma<<<dim3(gemm_blocks_g4), blk, 0, stream>>>(y0t, WTa, b1, xg, HH, G4);
    k_transpose<<<dim3(4096, 1), blk, 0, stream>>>(Whh1, WTh, G4, HH);
    k_lstm_rec<<<dim3(NWG_REC), blk, 0, stream>>>(xg, WTh, hbuf, y1t, ctr + 16,
                                                  hN + (size_t)BB * HH,
                                                  cN + (size_t)BB * HH);
    // ---- FC ----
    k_transpose<<<dim3(512, 1), blk, 0, stream>>>(fcw, WTfc, OO, HH);    // -> [H][O]
    k_gemm_wmma<<<dim3(gemm_blocks_fc), blk, 0, stream>>>(y1t, WTfc, fcb, fc, HH, OO);
}